// GHMCLoss_32469952758030
// MI455X (gfx1250) — compile-verified
//
#include <hip/hip_runtime.h>
#include <stdint.h>

#define BINS    10
#define BLOCK   256
#define MAXGRID 2048

#if defined(__has_builtin)
#  if __has_builtin(__builtin_amdgcn_global_load_async_to_lds_b128)
#    define HAVE_ASYNC_B128 1
#  endif
#  if __has_builtin(__builtin_amdgcn_s_wait_asynccnt)
#    define HAVE_WAIT_ASYNC 1
#  endif
#endif

#define AS1 __attribute__((address_space(1)))
#define AS3 __attribute__((address_space(3)))

typedef int v4i_t __attribute__((__vector_size__(16)));

#if defined(__HIP_DEVICE_COMPILE__)
#  if defined(HAVE_WAIT_ASYNC)
#    define WAIT_ASYNC(n) __builtin_amdgcn_s_wait_asynccnt(n)
#  else
#    define WAIT_ASYNC(n) asm volatile("s_wait_asynccnt %0" :: "n"(n) : "memory")
#  endif
#else
#  define WAIT_ASYNC(n) ((void)0)
#endif

// Async DMA of 16 bytes from global to LDS (ASYNCcnt-tracked, no VGPR return).
__device__ __forceinline__ void async_b128_to_lds(const float* gptr, void* lptr)
{
#if defined(__HIP_DEVICE_COMPILE__)
#  if defined(HAVE_ASYNC_B128)
    __builtin_amdgcn_global_load_async_to_lds_b128(
        (AS1 v4i_t*)gptr, (AS3 v4i_t*)lptr, 0, 0);
#  else
    uint32_t lds_off = (uint32_t)(uintptr_t)(AS3 void*)lptr;
    asm volatile("global_load_async_to_lds_b128 %0, %1, off"
                 :: "v"(lds_off), "v"(gptr) : "memory");
#  endif
#else
    (void)gptr; (void)lptr;
#endif
}

// Per-element update. With e = exp(-x), w1 = 1+e:
//   sigmoid(x) = 1/w1
//   bce = max(x,0) - x*t + log1p(exp(-|x|)) = x*(1-t) + log(w1)   (exact identity)
// Bin = clamp(floor(g*10), 0, 9) — matches searchsorted(edges,'right')-1.
__device__ __forceinline__ void ghm_accum(float x, float t, float2* hist, int tid)
{
    float e1 = __expf(-x);                 // v_exp_f32
    float w1 = 1.0f + e1;
#if defined(__HIP_DEVICE_COMPILE__)
    float s  = __builtin_amdgcn_rcpf(w1);  // v_rcp_f32; w1 >= 1 so safe
#else
    float s  = 1.0f / w1;
#endif
    float g  = fabsf(s - t);
    int   b  = (int)(g * 10.0f);
    b = (b > (BINS - 1)) ? (BINS - 1) : b;
    float bce = x * (1.0f - t) + __logf(w1);   // v_log_f32

    float2 h = hist[b * BLOCK + tid];          // ds_load_b64, conflict-free layout
    h.x += bce;
    h.y += 1.0f;
    hist[b * BLOCK + tid] = h;                 // ds_store_b64
}

extern "C" __global__ void __launch_bounds__(BLOCK)
ghm_hist_kernel(const float* __restrict__ pred,
                const float* __restrict__ target,
                float* __restrict__ partials,
                long long nvec, long long total)
{
    // Per-thread private histogram slots: {sum, count} as float2.
    // slot = b*256 + tid -> byte 2048*b + 8*tid: a wave's b64 ops hit all 64 banks once.
    __shared__ float2 hist[BINS * BLOCK];          // 20 KB
    __shared__ float4 pbuf[2][BLOCK];              // 8 KB  (async staging, per-lane slots)
    __shared__ float4 tbuf[2][BLOCK];              // 8 KB

    const int tid = threadIdx.x;

#pragma unroll
    for (int b = 0; b < BINS; ++b)
        hist[b * BLOCK + tid] = make_float2(0.0f, 0.0f);

    const long long i0     = (long long)blockIdx.x * BLOCK + tid;   // float4 units
    const long long stride = (long long)gridDim.x * BLOCK;

    if (i0 < nvec) {
        // Prologue: issue tile 0 (2 async ops outstanding).
        async_b128_to_lds(pred   + 4 * i0, &pbuf[0][tid]);
        async_b128_to_lds(target + 4 * i0, &tbuf[0][tid]);
        int par = 0;
        for (long long i = i0; i < nvec; i += stride) {
            long long inx = i + stride;
            long long ic  = (inx < nvec) ? inx : i;   // dummy re-issue keeps ASYNCcnt math uniform
            async_b128_to_lds(pred   + 4 * ic, &pbuf[par ^ 1][tid]);
            async_b128_to_lds(target + 4 * ic, &tbuf[par ^ 1][tid]);

            long long ipf = inx + stride;             // prefetch one further tile into L2
            if (ipf < nvec) {
                __builtin_prefetch(pred   + 4 * ipf, 0, 1);   // global_prefetch_b8
                __builtin_prefetch(target + 4 * ipf, 0, 1);
            }

            // 4 outstanding; async loads complete in order, so <=2 means tile i is in LDS.
            WAIT_ASYNC(2);
            asm volatile("" ::: "memory");

            float4 p = pbuf[par][tid];
            float4 t = tbuf[par][tid];
            ghm_accum(p.x, t.x, hist, tid);
            ghm_accum(p.y, t.y, hist, tid);
            ghm_accum(p.z, t.z, hist, tid);
            ghm_accum(p.w, t.w, hist, tid);
            par ^= 1;
        }
        WAIT_ASYNC(0);   // drain dangling dummy issues before LDS reuse below
    }

    // Scalar tail (total not divisible by 4).
    for (long long j = 4 * nvec + i0; j < total; j += stride)
        ghm_accum(pred[j], target[j], hist, tid);

    // Block reduction over the 256 thread-private columns.
    __syncthreads();
    for (int s = BLOCK / 2; s > 0; s >>= 1) {
        if (tid < s) {
#pragma unroll
            for (int b = 0; b < BINS; ++b) {
                float2 a = hist[b * BLOCK + tid];
                float2 c = hist[b * BLOCK + tid + s];
                a.x += c.x; a.y += c.y;
                hist[b * BLOCK + tid] = a;
            }
        }
        __syncthreads();
    }

    if (tid < BINS) {
        float2 r = hist[tid * BLOCK];
        partials[(long long)blockIdx.x * (2 * BINS) + tid]        = r.x;  // bce sum
        partials[(long long)blockIdx.x * (2 * BINS) + BINS + tid] = r.y;  // count
    }
}

extern "C" __global__ void __launch_bounds__(64)
ghm_final_kernel(const float* __restrict__ partials, int ngroups,
                 float* __restrict__ out)
{
    __shared__ float ssum[BINS];
    __shared__ float scnt[BINS];
    const int tid = threadIdx.x;

    if (tid < BINS) {
        float s = 0.0f, c = 0.0f;
        for (int i = 0; i < ngroups; ++i) {          // fixed order -> deterministic
            s += partials[i * (2 * BINS) + tid];
            c += partials[i * (2 * BINS) + BINS + tid];
        }
        ssum[tid] = s;
        scnt[tid] = c;
    }
    __syncthreads();

    if (tid == 0) {
        int n = 0;
        for (int b = 0; b < BINS; ++b) n += (scnt[b] > 0.0f) ? 1 : 0;
        float loss = 0.0f;
        if (n > 0) {
            float fn = (float)n;
            for (int b = 0; b < BINS; ++b)
                if (scnt[b] > 0.0f) loss += ssum[b] / (scnt[b] * fn);
        }
        out[0] = loss;   // LOSS_WEIGHT == 1, and tot cancels algebraically
    }
}

extern "C" void kernel_launch(void* const* d_in, const int* in_sizes, int n_in,
                              void* d_out, int out_size, void* d_ws, size_t ws_size,
                              hipStream_t stream)
{
    const float* pred   = (const float*)d_in[0];
    const float* target = (const float*)d_in[1];
    long long total = (long long)in_sizes[0];
    long long nvec  = total / 4;

    long long g = (nvec + BLOCK - 1) / BLOCK;
    if (g < 1) g = 1;
    if (g > MAXGRID) g = MAXGRID;
    int grid = (int)g;

    float* partials = (float*)d_ws;   // grid * 20 floats = 160 KB max

    ghm_hist_kernel<<<grid, BLOCK, 0, stream>>>(pred, target, partials, nvec, total);
    ghm_final_kernel<<<1, 64, 0, stream>>>(partials, grid, (float*)d_out);
}